// QuantumProjector_27401891348943
// MI455X (gfx1250) — compile-verified
//
#include <hip/hip_runtime.h>
#include <math.h>

typedef __attribute__((ext_vector_type(2))) float v2f;
typedef __attribute__((ext_vector_type(8))) float v8f;

#define BATCH 256
#define D_IN  1024
#define NQ    16
#define D_OUT 1024

// Fused: angles = x@W_pre^T + b_pre ; z = cos(qw)*cos(angles); zq = cumprod_q(z);
//        out = zq@W_post^T + b_post
// Quantum layer collapsed analytically: <Z_i> = prod_{j<=i} cos(w_j)*cos(angle_j)
// Grid: 16 blocks (one per 16-row batch tile), 256 threads = 8 waves.
__global__ __launch_bounds__(256) void qp_fused(
    const float* __restrict__ x,        // [256,1024]
    const float* __restrict__ W_pre,    // [16,1024]
    const float* __restrict__ b_pre,    // [16]
    const float* __restrict__ q_w,      // [16]
    const float* __restrict__ W_post,   // [1024,16]
    const float* __restrict__ b_post,   // [1024]
    float* __restrict__ out)            // [256,1024]
{
    __shared__ __align__(16) float part[8 * 256];  // 8 partial 16x16 tiles
    __shared__ __align__(16) float ztmp[256];      // z values (16 rows x 16 q)
    __shared__ __align__(16) float zq_s[256];      // cumprod result tile

    const int tid  = threadIdx.x;
    const int wave = tid >> 5;          // 0..7
    const int lane = tid & 31;          // 0..31
    const int half = lane >> 4;         // 0/1  (selects K-pair / M-upper-half)
    const int sub  = lane & 15;         // A row (m) or B col (n)
    const int m0   = blockIdx.x * 16;   // batch tile origin

    // ---------------- Stage 1: K-split GEMM1 via WMMA f32 16x16x4 ----------
    // Wave w reduces K range [w*128, w*128+128).
    {
        const float* aptr = x     + (m0 + sub) * D_IN + wave * 128 + 2 * half;
        const float* bptr = W_pre + sub        * D_IN + wave * 128 + 2 * half;

        v8f acc = {};
        #pragma unroll 4
        for (int k = 0; k < 128; k += 4) {
            v2f a = *(const v2f*)(aptr + k);   // A: rows m0+sub, K pair
            v2f b = *(const v2f*)(bptr + k);   // B: col sub (W_pre row), K pair
            acc = __builtin_amdgcn_wmma_f32_16x16x4_f32(
                false, a, false, b, (short)0, acc, false, false);
        }
        // C/D layout: VGPR r -> (m = r + 8*half, n = sub)
        #pragma unroll
        for (int r = 0; r < 8; ++r)
            part[wave * 256 + (r + 8 * half) * 16 + sub] = acc[r];
    }
    __syncthreads();

    // ---------------- Reduce partials, bias, cos, per-row cumprod ----------
    {
        // thread t owns element (m = t/16, q = t%16)
        const int q = tid & 15;
        float s = 0.f;
        #pragma unroll
        for (int w = 0; w < 8; ++w) s += part[w * 256 + tid];
        const float angle = s + b_pre[q];
        // |angle| ~ N(0,1) and |q_w| <= ~0.5: hardware V_COS_F32 fast path is
        // plenty accurate; avoids the Payne-Hanek slow-path code bloat.
        ztmp[tid] = __cosf(q_w[q]) * __cosf(angle);
    }
    __syncthreads();
    if (tid < 16) {                               // serial cumprod per row (16 elems)
        float prod = 1.f;
        #pragma unroll
        for (int q = 0; q < NQ; ++q) {
            prod *= ztmp[tid * 16 + q];
            zq_s[tid * 16 + q] = prod;            // zq[m, q] = prod_{j<=q} z_j
        }
    }
    __syncthreads();

    // ---------------- Stage 2: out tile = zq @ W_post^T + b_post -----------
    // A operand (zq tile, 16x16, K=16) loaded once from LDS per wave.
    v2f a0 = *(const v2f*)(&zq_s[sub * 16 +  0 + 2 * half]);
    v2f a1 = *(const v2f*)(&zq_s[sub * 16 +  4 + 2 * half]);
    v2f a2 = *(const v2f*)(&zq_s[sub * 16 +  8 + 2 * half]);
    v2f a3 = *(const v2f*)(&zq_s[sub * 16 + 12 + 2 * half]);

    #pragma unroll
    for (int j = 0; j < 8; ++j) {
        const int n0 = (wave * 8 + j) * 16;       // output column tile
        const float* bptr = W_post + (n0 + sub) * NQ + 2 * half;

        v2f b0 = *(const v2f*)(bptr +  0);
        v2f b1 = *(const v2f*)(bptr +  4);
        v2f b2 = *(const v2f*)(bptr +  8);
        v2f b3 = *(const v2f*)(bptr + 12);

        v8f acc = {};
        acc = __builtin_amdgcn_wmma_f32_16x16x4_f32(false, a0, false, b0, (short)0, acc, false, false);
        acc = __builtin_amdgcn_wmma_f32_16x16x4_f32(false, a1, false, b1, (short)0, acc, false, false);
        acc = __builtin_amdgcn_wmma_f32_16x16x4_f32(false, a2, false, b2, (short)0, acc, false, false);
        acc = __builtin_amdgcn_wmma_f32_16x16x4_f32(false, a3, false, b3, (short)0, acc, false, false);

        const float bias = b_post[n0 + sub];
        #pragma unroll
        for (int r = 0; r < 8; ++r)
            out[(m0 + r + 8 * half) * D_OUT + n0 + sub] = acc[r] + bias;
    }
}

extern "C" void kernel_launch(void* const* d_in, const int* in_sizes, int n_in,
                              void* d_out, int out_size, void* d_ws, size_t ws_size,
                              hipStream_t stream) {
    const float* x      = (const float*)d_in[0];
    const float* W_pre  = (const float*)d_in[1];
    const float* b_pre  = (const float*)d_in[2];
    const float* q_w    = (const float*)d_in[3];
    const float* W_post = (const float*)d_in[4];
    const float* b_post = (const float*)d_in[5];
    float* out = (float*)d_out;

    qp_fused<<<dim3(BATCH / 16), dim3(256), 0, stream>>>(
        x, W_pre, b_pre, q_w, W_post, b_post, out);
}